// RGCNEncoder_49761491091971
// MI455X (gfx1250) — compile-verified
//
#include <hip/hip_runtime.h>

#define D 128
#define NREL 16
#define PRD 288  // dwords per LDS pair-row (64 pair-rows): conflict-free b64 consume

typedef __attribute__((ext_vector_type(2))) float v2f;
typedef __attribute__((ext_vector_type(8))) float v8f;

// ---------------- utility: zero a float4 region (agg / cnt init each call) ----------
__global__ __launch_bounds__(256) void zero_f4_kernel(float4* __restrict__ p, long n4) {
  long i = (long)blockIdx.x * blockDim.x + threadIdx.x;
  long stride = (long)gridDim.x * blockDim.x;
  float4 z; z.x = z.y = z.z = z.w = 0.0f;
  for (; i < n4; i += stride) p[i] = z;
}

// ---------------- per-(dst,rel) edge counts (shared by both layers) -----------------
__global__ __launch_bounds__(256) void count_kernel(const int* __restrict__ dst,
                                                    const int* __restrict__ etype,
                                                    unsigned* __restrict__ cnt, int nE) {
  int e = blockIdx.x * 256 + threadIdx.x;
  if (e < nE) atomicAdd(&cnt[(long)dst[e] * NREL + etype[e]], 1u);
}

__global__ __launch_bounds__(256) void inv_kernel(const unsigned* __restrict__ cnt,
                                                  float* __restrict__ inv, int n) {
  int i = blockIdx.x * 256 + threadIdx.x;
  if (i < n) {
    unsigned c = cnt[i];
    inv[i] = 1.0f / (float)(c > 1u ? c : 1u);  // mean with max(cnt,1)
  }
}

// ---------------- edge scatter: one wave per edge, 128 floats = 4/lane --------------
__global__ __launch_bounds__(256) void scatter_kernel(const int* __restrict__ src,
                                                      const int* __restrict__ dst,
                                                      const int* __restrict__ etype,
                                                      const float* __restrict__ x,
                                                      float* __restrict__ agg, int nE) {
  int e = blockIdx.x * 8 + (threadIdx.x >> 5);
  int lane = threadIdx.x & 31;
  if (e >= nE) return;
  int s = src[e], d = dst[e], t = etype[e];
  float4 v = ((const float4*)(x + (long)s * D))[lane];
  float* a = agg + ((long)d * NREL + t) * D + lane * 4;
  atomicAdd(a + 0, v.x);
  atomicAdd(a + 1, v.y);
  atomicAdd(a + 2, v.z);
  atomicAdd(a + 3, v.w);
}

// ---------------- fused RGCN layer GEMM with V_WMMA_F32_16X16X4_F32 -----------------
// out[rows,128] = relu?( bias + x@root + sum_r (inv(row,r) * agg[row,r,:]) @ W[r] )
// Block = 128 threads (4 waves). Each wave owns TWO 16-row tiles (32 rows) and all
// 128 output columns: agg is read once from HBM, and every B ds_load_b64 feeds two
// WMMAs. B staged per-matrix in LDS, pair-interleaved: element (k,n) at dword
// (k>>1)*PRD + n*2 + (k&1), so each lane's (B[kk][n], B[kk+1][n]) is one aligned
// 8-byte ds_load_b64 (no VGPR pairing moves), conflict-free across both lane halves.
__global__ __launch_bounds__(128) void rgcn_gemm_kernel(
    const float* __restrict__ x, const float* __restrict__ agg,
    const float* __restrict__ inv, const float* __restrict__ W,
    const float* __restrict__ root, const float* __restrict__ bias,
    float* __restrict__ out, int nNodes, int doRelu) {
  extern __shared__ float sB[];  // 64 * PRD dwords = 73728 B
  const int lane = threadIdx.x & 31;
  const int wave = threadIdx.x >> 5;
  const int l15 = lane & 15;
  const int halfk = (lane >> 4) * 2;  // K sub-offset for this lane half (0 or 2)
  const int hi8 = (lane >> 4) * 8;    // row offset in C/D layout
  const int tile0 = blockIdx.x * 8 + wave * 2;
  const int tile1 = tile0 + 1;
  const bool act0 = tile0 * 16 < nNodes;  // wave-uniform => EXEC stays all-1
  const bool act1 = tile1 * 16 < nNodes;
  const int rb0 = act0 ? tile0 * 16 : 0;  // clamp inactive tiles to valid rows
  const int rb1 = act1 ? tile1 * 16 : rb0;
  const long row0 = (long)rb0 + l15;
  const long row1 = (long)rb1 + l15;

  v8f acc0[8], acc1[8];
#pragma unroll
  for (int cb = 0; cb < 8; ++cb) {
    float bv = bias[cb * 16 + l15];
#pragma unroll
    for (int v = 0; v < 8; ++v) { acc0[cb][v] = bv; acc1[cb][v] = bv; }
  }

  for (int m = 0; m < NREL + 1; ++m) {  // 16 relations + root
    const float* src = (m < NREL) ? (W + (long)m * D * D) : root;
    __syncthreads();
    // cooperative stage of B into pair-interleaved LDS; global float4 coalesced
    for (int i = threadIdx.x; i < 2048; i += 128) {
      int t = i >> 5, n4 = i & 31;
      float4 lo = ((const float4*)(src + (2 * t) * D))[n4];
      float4 hi = ((const float4*)(src + (2 * t + 1) * D))[n4];
      float* dstp = sB + t * PRD + n4 * 8;
      *(float4*)(dstp) = make_float4(lo.x, hi.x, lo.y, hi.y);
      *(float4*)(dstp + 4) = make_float4(lo.z, hi.z, lo.w, hi.w);
    }
    __syncthreads();

    const float *A0, *A1;
    float s0, s1;
    if (m < NREL) {
      A0 = agg + (row0 * NREL + m) * D;  s0 = inv[row0 * NREL + m];
      A1 = agg + (row1 * NREL + m) * D;  s1 = inv[row1 * NREL + m];
    } else {
      A0 = x + row0 * D;  s0 = 1.0f;
      A1 = x + row1 * D;  s1 = 1.0f;
    }
#pragma unroll 2
    for (int k0 = 0; k0 < D; k0 += 4) {
      int kk = k0 + halfk;
      v2f a0 = *(const v2f*)(A0 + kk);  // A 16x4 f32 lane layout (ISA 7.12.2)
      a0.x *= s0; a0.y *= s0;
      v2f a1 = *(const v2f*)(A1 + kk);
      a1.x *= s1; a1.y *= s1;
      const float* bp = sB + (kk >> 1) * PRD + l15 * 2;
#pragma unroll
      for (int cb = 0; cb < 8; ++cb) {
        v2f b = *(const v2f*)(bp + cb * 32);  // one ds_load_b64, aligned pair
        acc0[cb] = __builtin_amdgcn_wmma_f32_16x16x4_f32(
            false, a0, false, b, (short)0, acc0[cb], false, false);
        acc1[cb] = __builtin_amdgcn_wmma_f32_16x16x4_f32(
            false, a1, false, b, (short)0, acc1[cb], false, false);
      }
    }
  }

  if (act0) {
#pragma unroll
    for (int cb = 0; cb < 8; ++cb)
#pragma unroll
      for (int v = 0; v < 8; ++v) {
        float val = acc0[cb][v];
        if (doRelu) val = fmaxf(val, 0.0f);
        out[(long)(rb0 + v + hi8) * D + cb * 16 + l15] = val;
      }
  }
  if (act1) {
#pragma unroll
    for (int cb = 0; cb < 8; ++cb)
#pragma unroll
      for (int v = 0; v < 8; ++v) {
        float val = acc1[cb][v];
        if (doRelu) val = fmaxf(val, 0.0f);
        out[(long)(rb1 + v + hi8) * D + cb * 16 + l15] = val;
      }
  }
}

extern "C" void kernel_launch(void* const* d_in, const int* in_sizes, int n_in,
                              void* d_out, int out_size, void* d_ws, size_t ws_size,
                              hipStream_t stream) {
  (void)n_in; (void)out_size; (void)ws_size;
  const int E = in_sizes[1];                  // edge_type count = #edges
  const int* eSrc = (const int*)d_in[0];      // edge_index[0, :]
  const int* eDst = eSrc + E;                 // edge_index[1, :]
  const int* eType = (const int*)d_in[1];
  const float* emb = (const float*)d_in[2];
  const float* W1 = (const float*)d_in[3];
  const float* root1 = (const float*)d_in[4];
  const float* b1 = (const float*)d_in[5];
  const float* W2 = (const float*)d_in[6];
  const float* root2 = (const float*)d_in[7];
  const float* b2 = (const float*)d_in[8];
  float* out = (float*)d_out;
  const int N = in_sizes[2] / D;              // 50000
  const int segs = N * NREL;                  // 800000

  // workspace layout: agg | cnt | inv | x1
  char* ws = (char*)d_ws;
  float* agg = (float*)ws;
  size_t aggBytes = (size_t)segs * D * sizeof(float);        // 409.6 MB
  unsigned* cnt = (unsigned*)(ws + aggBytes);
  float* inv = (float*)(ws + aggBytes + (size_t)segs * 4);
  float* x1 = (float*)(ws + aggBytes + 2 * (size_t)segs * 4);

  const size_t ldsB = (size_t)64 * PRD * sizeof(float);      // 73728 B
  const int nTiles = (N + 15) / 16;                          // 3125
  const int gemmBlocks = (nTiles + 7) / 8;                   // 8 tiles / block

  // shared counts
  zero_f4_kernel<<<1024, 256, 0, stream>>>((float4*)cnt, (long)segs / 4);
  count_kernel<<<(E + 255) / 256, 256, 0, stream>>>(eDst, eType, cnt, E);
  inv_kernel<<<(segs + 255) / 256, 256, 0, stream>>>(cnt, inv, segs);

  // layer 1: emb -> x1 (relu)
  zero_f4_kernel<<<4096, 256, 0, stream>>>((float4*)agg, (long)segs * D / 4);
  scatter_kernel<<<(E + 7) / 8, 256, 0, stream>>>(eSrc, eDst, eType, emb, agg, E);
  rgcn_gemm_kernel<<<gemmBlocks, 128, ldsB, stream>>>(emb, agg, inv, W1, root1, b1,
                                                      x1, N, 1);

  // layer 2: x1 -> out
  zero_f4_kernel<<<4096, 256, 0, stream>>>((float4*)agg, (long)segs * D / 4);
  scatter_kernel<<<(E + 7) / 8, 256, 0, stream>>>(eSrc, eDst, eType, x1, agg, E);
  rgcn_gemm_kernel<<<gemmBlocks, 128, ldsB, stream>>>(x1, agg, inv, W2, root2, b2,
                                                      out, N, 0);
}